// PeriodicTemporalStateAdapter_51299089383907
// MI455X (gfx1250) — compile-verified
//
#include <hip/hip_runtime.h>
#include <math.h>

#define NB   8
#define C_CH 256
#define SEQ  64
#define HW   256

typedef float v2f __attribute__((ext_vector_type(2)));
typedef float v8f __attribute__((ext_vector_type(8)));

__device__ __forceinline__ float gelu_f(float v) {
    return 0.5f * v * (1.0f + erff(v * 0.70710678118654752440f));
}
__device__ __forceinline__ float sigmoid_f(float v) {
    return 1.0f / (1.0f + expf(-v));
}

// ---------------------------------------------------------------------------
// Generic fp32 WMMA GEMM:  Out[m][o] = epi( alpha * sum_{k,t} A(m+shift(t), k) * W[o][k][t] )
// One wave (32 lanes) computes one 16x16 tile using V_WMMA_F32_16X16X4_F32.
// A addressing: m -> (an = m / a_hdiv, ar = m % a_hdiv); elem = A[an*a_ns + ar*a_rs + k*a_ks]
// taps>1: temporal conv along ar with zero padding (ar + t - taps/2 must be in [0, a_hdiv)).
// Out addressing: same scheme with o_* params (lets us store sgate transposed).
// epi: 0=identity  1=bn+gelu(esc,esh)  2=bias+sigmoid(esh)  3=bn+sigmoid(esc,esh)
// Fast path (wave-uniform branch): a_ks==1, taps==1, K%8==0 -> contiguous v2f
// loads (global_load_b64) and 2 WMMAs per loop trip, no masking.
// ---------------------------------------------------------------------------
__global__ void gemm_wmma_f32(
    const float* __restrict__ A, const float* __restrict__ W,
    const float* __restrict__ esc, const float* __restrict__ esh,
    float* __restrict__ Out,
    int M, int Nout, int K, int taps,
    int a_hdiv, long a_ns, int a_rs, int a_ks,
    int o_hdiv, long o_ns, int o_rs, int o_cs,
    float alpha, int epi, int accum)
{
    int gwave = (int)((blockIdx.x * blockDim.x + threadIdx.x) >> 5);
    int lane  = threadIdx.x & 31;
    int tilesN = Nout >> 4;
    int tilesM = M >> 4;
    if (gwave >= tilesM * tilesN) return;   // wave-uniform exit (EXEC stays full)

    int tm = (gwave / tilesN) << 4;
    int to = (gwave % tilesN) << 4;
    int l15  = lane & 15;
    int hi   = lane >> 4;          // 0: K 0,1   1: K 2,3  (f32 WMMA A/B lane layout)
    int koff = hi << 1;

    int arow = tm + l15;
    int an = arow / a_hdiv;
    int ar = arow - an * a_hdiv;
    int bcol = to + l15;
    long wrow = (long)bcol * (long)K * taps;
    int half = taps >> 1;

    v8f acc = {0.f,0.f,0.f,0.f,0.f,0.f,0.f,0.f};

    if (a_ks == 1 && taps == 1 && (K & 7) == 0) {
        // ---- fast path: contiguous 8-byte loads, 2 WMMAs per iteration ----
        const float* arp = A + (long)an * a_ns + (long)ar * a_rs + koff;
        const float* wrp = W + wrow + koff;
        for (int kc = 0; kc < K; kc += 8) {
            v2f a0 = *(const v2f*)(arp + kc);
            v2f b0 = *(const v2f*)(wrp + kc);
            v2f a1 = *(const v2f*)(arp + kc + 4);
            v2f b1 = *(const v2f*)(wrp + kc + 4);
            acc = __builtin_amdgcn_wmma_f32_16x16x4_f32(
                false, a0, false, b0, (short)0, acc, false, false);
            acc = __builtin_amdgcn_wmma_f32_16x16x4_f32(
                false, a1, false, b1, (short)0, acc, false, false);
        }
    } else {
        // ---- generic path: tapped conv / strided A / ragged K ----
        for (int t = 0; t < taps; ++t) {
            int rr = ar + t - half;
            bool rowok = (rr >= 0) && (rr < a_hdiv);
            int rc = rr < 0 ? 0 : (rr >= a_hdiv ? a_hdiv - 1 : rr);
            const float* arp = A + (long)an * a_ns + (long)rc * a_rs;
            const float* wrp = W + wrow + t;
            for (int kc = 0; kc < K; kc += 4) {
                int k0 = kc + koff;
                int k1 = k0 + 1;
                bool k0ok = k0 < K, k1ok = k1 < K;
                int k0c = k0ok ? k0 : 0;
                int k1c = k1ok ? k1 : 0;
                float a0 = arp[(long)k0c * a_ks];
                float a1 = arp[(long)k1c * a_ks];
                float b0 = wrp[(long)k0c * taps];
                float b1 = wrp[(long)k1c * taps];
                v2f a, b;
                a.x = (rowok && k0ok) ? a0 : 0.0f;
                a.y = (rowok && k1ok) ? a1 : 0.0f;
                b.x = k0ok ? b0 : 0.0f;
                b.y = k1ok ? b1 : 0.0f;
                acc = __builtin_amdgcn_wmma_f32_16x16x4_f32(
                    false, a, false, b, (short)0, acc, false, false);
            }
        }
    }

    int col = to + l15;
    int rbase = tm + (hi << 3);    // D layout: lanes 0-15 rows 0-7, lanes 16-31 rows 8-15
    float sc = 1.0f, sh = 0.0f;
    if (epi == 1 || epi == 3) { sc = esc[col]; sh = esh[col]; }
    else if (epi == 2)        { sh = esh[col]; }
#pragma unroll
    for (int r = 0; r < 8; ++r) {
        int m = rbase + r;
        float v = acc[r] * alpha;
        if      (epi == 1) v = gelu_f(v * sc + sh);
        else if (epi == 2) v = sigmoid_f(v + sh);
        else if (epi == 3) v = sigmoid_f(v * sc + sh);
        int on = m / o_hdiv;
        int orr = m - on * o_hdiv;
        long oa = (long)on * o_ns + (long)orr * o_rs + (long)col * o_cs;
        if (accum) Out[oa] += v; else Out[oa] = v;
    }
}

// ---------------------------------------------------------------------------
// Pass 1 over x: desc[n][c][s] = mean_{hw} x ; sp[n][c][hw] = mean_s x
// ---------------------------------------------------------------------------
__global__ void reduce_kernel(const float* __restrict__ x,
                              float* __restrict__ desc, float* __restrict__ sp)
{
    int nc = blockIdx.x;          // n*256 + c
    int t  = threadIdx.x;         // hw
    __shared__ float red[256];
    const float* base = x + (long)nc * SEQ * HW;
    float acc = 0.f;
    for (int s = 0; s < SEQ; ++s) {
        float v = base[s * HW + t];
        int sp8 = s + 8 < SEQ ? s + 8 : SEQ - 1;
        __builtin_prefetch(base + sp8 * HW + t, 0, 0);   // global_prefetch_b8
        acc += v;
        red[t] = v; __syncthreads();
        for (int o = 128; o > 0; o >>= 1) {
            if (t < o) red[t] += red[t + o];
            __syncthreads();
        }
        if (t == 0) desc[(long)nc * SEQ + s] = red[0] * (1.0f / 256.0f);
        __syncthreads();
    }
    sp[(long)nc * HW + t] = acc * (1.0f / 64.0f);
}

// LayerNorm over channels per token -> dnT[(n*64+s)][c]
__global__ void ln_kernel(const float* __restrict__ desc,
                          const float* __restrict__ g, const float* __restrict__ b,
                          float* __restrict__ dnT)
{
    int tok = blockIdx.x;               // n*64 + s
    int n = tok >> 6, s = tok & 63;
    int c = threadIdx.x;
    __shared__ float r1[256], r2[256];
    float v = desc[((long)(n * C_CH + c)) * SEQ + s];
    r1[c] = v; r2[c] = v * v; __syncthreads();
    for (int o = 128; o > 0; o >>= 1) {
        if (c < o) { r1[c] += r1[c + o]; r2[c] += r2[c + o]; }
        __syncthreads();
    }
    float mu  = r1[0] * (1.0f / 256.0f);
    float var = r2[0] * (1.0f / 256.0f) - mu * mu;
    dnT[(long)tok * C_CH + c] = (v - mu) * rsqrtf(var + 1e-5f) * g[c] + b[c];
}

// mi = [dn, diff1, diff2] (3C), gi = [dn, |diff1|] (2C)
__global__ void buildmi_kernel(const float* __restrict__ dnT,
                               float* __restrict__ mi, float* __restrict__ gi)
{
    int tok = blockIdx.x; int s = tok & 63; int c = threadIdx.x;
    float d0 = dnT[(long)tok * C_CH + c];
    float d1 = (s >= 1) ? d0 - dnT[(long)(tok - 1) * C_CH + c] : 0.f;
    float d2 = (s >= 2) ? d0 - dnT[(long)(tok - 2) * C_CH + c] : 0.f;
    long mb = (long)tok * 768;
    mi[mb + c] = d0; mi[mb + 256 + c] = d1; mi[mb + 512 + c] = d2;
    long gb = (long)tok * 512;
    gi[gb + c] = d0; gi[gb + 256 + c] = fabsf(d1);
}

__global__ void zero_kernel(float* __restrict__ p, int n) {
    int i = blockIdx.x * 256 + threadIdx.x;
    if (i < n) p[i] = 0.f;
}

// 64-point ortho rDFT per (n,c); accumulate |F| and complex sum over channels
__global__ void dft_kernel(const float* __restrict__ dnT,
                           float* __restrict__ fe, float* __restrict__ sre,
                           float* __restrict__ sim)
{
    int nc = blockIdx.x; int n = nc >> 8; int c = nc & 255;
    int f = threadIdx.x;
    if (f >= 33) return;
    float re = 0.f, im = 0.f;
    for (int s = 0; s < SEQ; ++s) {
        float v = dnT[((long)(n << 6) + s) * C_CH + c];
        float th = 6.283185307179586f * (float)(f * s) / 64.0f;
        float sv, cv; sincosf(th, &sv, &cv);
        re += v * cv; im -= v * sv;
    }
    re *= 0.125f; im *= 0.125f;   // 1/sqrt(64)
    atomicAdd(&fe[n * 33 + f], sqrtf(re * re + im * im));
    atomicAdd(&sre[n * 33 + f], re);
    atomicAdd(&sim[n * 33 + f], im);
}

// Peak pick, phase features pf[(n,s)][3] = {cos, sin, conf}; nst[n*8+0..1] = conf, peak_norm
__global__ void peak_kernel(const float* __restrict__ fe, const float* __restrict__ sre,
                            const float* __restrict__ sim,
                            float* __restrict__ pf, float* __restrict__ nst)
{
    int n = blockIdx.x; int t = threadIdx.x;
    __shared__ float sh[3];
    if (t == 0) {
        const int low = 7;                 // round(33*0.2)=7, clamped to [1,32]
        float best = -1.f; int bi = low; float hsum = 0.f;
        for (int f = low; f < 33; ++f) {
            float v = fe[n * 33 + f];
            hsum += v;
            if (v > best) { best = v; bi = f; }
        }
        float conf = best / fmaxf(hsum, 1e-6f);
        float ph = atan2f(sim[n * 33 + bi], sre[n * 33 + bi]);
        sh[0] = 6.283185307179586f / 64.0f * (float)bi;
        sh[1] = ph; sh[2] = conf;
        nst[n * 8 + 0] = conf;
        nst[n * 8 + 1] = (float)bi / 32.0f;
    }
    __syncthreads();
    if (t < SEQ) {
        float arg = sh[0] * (float)t + sh[1];
        long r = ((long)n * SEQ + t) * 3;
        pf[r + 0] = cosf(arg); pf[r + 1] = sinf(arg); pf[r + 2] = sh[2];
    }
}

__global__ void mul_kernel(const float* __restrict__ a, const float* __restrict__ b,
                           float* __restrict__ o)
{
    long i = (long)blockIdx.x * 256 + threadIdx.x;
    o[i] = a[i] * b[i];
}

// Gated fwd/bwd scans per (n,c); fill cat3 = [dn, fwd_mem, bwd_mem]
__global__ void scan_kernel(const float* __restrict__ dnT, const float* __restrict__ gate,
                            float* __restrict__ cat3)
{
    int n = blockIdx.x; int c = threadIdx.x;
    long t0 = (long)n * SEQ;
    float st = dnT[t0 * C_CH + c];
    cat3[t0 * 768 + 256 + c] = st;
    for (int s = 1; s < SEQ; ++s) {
        long tk = t0 + s;
        float g = gate[tk * C_CH + c];
        float d = dnT[tk * C_CH + c];
        st = g * st + (1.0f - g) * d;
        cat3[tk * 768 + 256 + c] = st;
    }
    st = dnT[(t0 + SEQ - 1) * C_CH + c];
    cat3[(t0 + SEQ - 1) * 768 + 512 + c] = st;
    for (int s = SEQ - 2; s >= 0; --s) {
        long tk = t0 + s;
        float g = gate[tk * C_CH + c];
        float d = dnT[tk * C_CH + c];
        st = g * st + (1.0f - g) * d;
        cat3[tk * 768 + 512 + c] = st;
    }
    for (int s = 0; s < SEQ; ++s)
        cat3[(t0 + s) * 768 + c] = dnT[(t0 + s) * C_CH + c];
}

// Per-batch stats + router MLP (261 -> 64 -> 3 -> softmax) ; writes nst[n*8+2..4]
__global__ void stats_router_kernel(const float* __restrict__ dnT, const float* __restrict__ mi,
                                    const float* __restrict__ motion, const float* __restrict__ memory,
                                    const float* __restrict__ rtw1, const float* __restrict__ rtb1,
                                    const float* __restrict__ rtw2, const float* __restrict__ rtb2,
                                    float* __restrict__ nst)
{
    int n = blockIdx.x; int c = threadIdx.x;
    __shared__ float dnm[256], red[256], stats[5], hdn[64];
    float sd = 0.f, sd2 = 0.f, adiff = 0.f, amot = 0.f, amem = 0.f;
    for (int s = 0; s < SEQ; ++s) {
        long tk = (long)n * SEQ + s;
        float d = dnT[tk * C_CH + c];
        sd += d; sd2 += d * d;
        adiff += fabsf(mi[tk * 768 + 256 + c]);
        amot  += fabsf(motion[tk * C_CH + c]);
        amem  += fabsf(memory[tk * C_CH + c] - d);
    }
    float mean = sd * (1.0f / 64.0f);
    dnm[c] = mean;
    float stdc = sqrtf(fmaxf(sd2 * (1.0f / 64.0f) - mean * mean, 0.f));

    float sums[4]; float vals[4] = {stdc, adiff, amot, amem};
    for (int q = 0; q < 4; ++q) {
        red[c] = vals[q]; __syncthreads();
        for (int o = 128; o > 0; o >>= 1) {
            if (c < o) red[c] += red[c + o];
            __syncthreads();
        }
        sums[q] = red[0]; __syncthreads();
    }
    if (c == 0) {
        stats[0] = sums[0] * (1.0f / 256.0f);
        stats[1] = sums[1] * (1.0f / 16384.0f);
        stats[2] = nst[n * 8 + 0];
        stats[3] = nst[n * 8 + 1];
        stats[4] = (sums[3] + sums[2]) * (1.0f / 16384.0f);
    }
    __syncthreads();
    if (c < 64) {
        float a = rtb1[c];
        const float* wrow = rtw1 + (long)c * 261;
        for (int k = 0; k < 256; ++k) a += dnm[k] * wrow[k];
        for (int i = 0; i < 5; ++i)  a += stats[i] * wrow[256 + i];
        hdn[c] = gelu_f(a);
    }
    __syncthreads();
    if (c == 0) {
        float lg[3];
        float mx = -1e30f;
        for (int q = 0; q < 3; ++q) {
            float a = rtb2[q];
            for (int j = 0; j < 64; ++j) a += hdn[j] * rtw2[q * 64 + j];
            lg[q] = a; if (a > mx) mx = a;
        }
        float e0 = expf(lg[0] - mx), e1 = expf(lg[1] - mx), e2 = expf(lg[2] - mx);
        float inv = 1.0f / (e0 + e1 + e2);
        nst[n * 8 + 2] = e0 * inv; nst[n * 8 + 3] = e1 * inv; nst[n * 8 + 4] = e2 * inv;
    }
}

// fused + cat4 = [dn, fused, motion-phase, memory-dn]
__global__ void cat4_kernel(const float* __restrict__ dnT, const float* __restrict__ motion,
                            const float* __restrict__ phase, const float* __restrict__ memory,
                            const float* __restrict__ nst, float* __restrict__ cat4)
{
    int tok = blockIdx.x; int n = tok >> 6; int c = threadIdx.x;
    float b0 = nst[n * 8 + 2], b1 = nst[n * 8 + 3], b2 = nst[n * 8 + 4];
    long i = (long)tok * C_CH + c;
    float d = dnT[i], mo = motion[i], ph = phase[i], me = memory[i];
    float fu = b0 * mo + b1 * ph + b2 * me;
    long ob = (long)tok * 1024;
    cat4[ob + c] = d; cat4[ob + 256 + c] = fu;
    cat4[ob + 512 + c] = mo - ph; cat4[ob + 768 + c] = me - d;
}

// dtok = delta0 * sigmoid-gate * res_scale[c]
__global__ void dtok_kernel(const float* __restrict__ d0, const float* __restrict__ fg,
                            const float* __restrict__ rsc, float* __restrict__ dtok)
{
    int c = threadIdx.x;
    long i = (long)blockIdx.x * C_CH + c;
    dtok[i] = d0[i] * fg[i] * rsc[c];
}

// out[n,c,s,h,w] = x + dtok[n,s,c] * sgate[n,c,hw]   (sgate stored [n][c][hw])
__global__ void final_kernel(const float* __restrict__ x, const float* __restrict__ dtok,
                             const float* __restrict__ sgate, float* __restrict__ out)
{
    long b = blockIdx.x;                 // (n*256+c)*64 + s
    int hw = threadIdx.x;
    int s = (int)(b & 63);
    long nc = b >> 6;
    int c = (int)(nc & 255);
    int n = (int)(nc >> 8);
    float d = dtok[(((long)n * SEQ + s) << 8) + c];
    float g = sgate[(nc << 8) + hw];
    long idx = (b << 8) + hw;
    out[idx] = x[idx] + d * g;
}

// ---------------------------------------------------------------------------
static inline void launch_gemm(hipStream_t st, const float* A, const float* W,
    const float* esc, const float* esh, float* Out,
    int M, int N, int K, int taps,
    int a_hdiv, long a_ns, int a_rs, int a_ks,
    int o_hdiv, long o_ns, int o_rs, int o_cs,
    float alpha, int epi, int accum)
{
    int tiles = (M / 16) * (N / 16);
    int blocks = (tiles + 3) / 4;
    gemm_wmma_f32<<<blocks, 128, 0, st>>>(A, W, esc, esh, Out, M, N, K, taps,
        a_hdiv, a_ns, a_rs, a_ks, o_hdiv, o_ns, o_rs, o_cs, alpha, epi, accum);
}

extern "C" void kernel_launch(void* const* d_in, const int* in_sizes, int n_in,
                              void* d_out, int out_size, void* d_ws, size_t ws_size,
                              hipStream_t stream)
{
    (void)in_sizes; (void)n_in; (void)out_size; (void)ws_size;
    const float* x     = (const float*)d_in[0];
    const float* ln_g  = (const float*)d_in[1];
    const float* ln_b  = (const float*)d_in[2];
    const float* mb_w3 = (const float*)d_in[3];
    const float* mb_s3 = (const float*)d_in[4];
    const float* mb_b3 = (const float*)d_in[5];
    const float* mb_w5 = (const float*)d_in[6];
    const float* mb_s5 = (const float*)d_in[7];
    const float* mb_b5 = (const float*)d_in[8];
    const float* mb_w7 = (const float*)d_in[9];
    const float* mb_s7 = (const float*)d_in[10];
    const float* mb_b7 = (const float*)d_in[11];
    const float* mf_w  = (const float*)d_in[12];
    const float* mf_s  = (const float*)d_in[13];
    const float* mf_b  = (const float*)d_in[14];
    const float* pg_w1 = (const float*)d_in[15];
    const float* pg_s1 = (const float*)d_in[16];
    const float* pg_b1 = (const float*)d_in[17];
    const float* pg_w2 = (const float*)d_in[18];
    const float* pg_b2 = (const float*)d_in[19];
    const float* pp_w  = (const float*)d_in[20];
    const float* pp_s  = (const float*)d_in[21];
    const float* pp_b  = (const float*)d_in[22];
    const float* mg_w  = (const float*)d_in[23];
    const float* mg_b  = (const float*)d_in[24];
    const float* mo_w1 = (const float*)d_in[25];
    const float* mo_s1 = (const float*)d_in[26];
    const float* mo_b1 = (const float*)d_in[27];
    const float* mo_w2 = (const float*)d_in[28];
    const float* rt_w1 = (const float*)d_in[29];
    const float* rt_b1 = (const float*)d_in[30];
    const float* rt_w2 = (const float*)d_in[31];
    const float* rt_b2 = (const float*)d_in[32];
    const float* bp_w1 = (const float*)d_in[33];
    const float* bp_s1 = (const float*)d_in[34];
    const float* bp_b1 = (const float*)d_in[35];
    const float* bp_w2 = (const float*)d_in[36];
    const float* res_s = (const float*)d_in[37];
    const float* fg_w  = (const float*)d_in[38];
    const float* fg_b  = (const float*)d_in[39];
    const float* sg_w  = (const float*)d_in[40];
    const float* sg_s  = (const float*)d_in[41];
    const float* sg_b  = (const float*)d_in[42];
    float* out = (float*)d_out;

    float* w = (float*)d_ws;
    size_t off = 0;
    auto alloc = [&](size_t n) { float* p = w + off; off += n; return p; };
    float* desc   = alloc(131072);   // (N,C,S)
    float* sp     = alloc(524288);   // (N,C,HW)
    float* dnT    = alloc(131072);   // (512,256)
    float* mi     = alloc(393216);   // (512,768)
    float* gi     = alloc(262144);   // (512,512)
    float* motA   = alloc(131072);
    float* motion = alloc(131072);
    float* fe     = alloc(512);
    float* sre    = alloc(512);
    float* sim    = alloc(512);
    float* pf     = alloc(2048);     // (512,3) padded
    float* nst    = alloc(64);       // per-n: conf, peak_norm, bw0..2
    float* pgh    = alloc(32768);    // (512,64)
    float* pgate  = alloc(131072);
    float* dnp    = alloc(131072);
    float* phase  = alloc(131072);
    float* gate   = alloc(131072);
    float* cat3   = alloc(393216);
    float* memh   = alloc(131072);
    float* memory = alloc(131072);
    float* cat4   = alloc(524288);
    float* bph    = alloc(131072);
    float* delta0 = alloc(131072);
    float* fgate  = alloc(131072);
    float* dtok   = alloc(131072);
    float* sgate  = alloc(524288);   // (N,C,HW)

    const int M = NB * SEQ;          // 512 tokens

    // 1) x reductions
    reduce_kernel<<<NB * C_CH, 256, 0, stream>>>(x, desc, sp);
    // 2) LayerNorm -> dnT
    ln_kernel<<<M, 256, 0, stream>>>(desc, ln_g, ln_b, dnT);
    // 3) diffs / concats
    buildmi_kernel<<<M, 256, 0, stream>>>(dnT, mi, gi);
    // 4) motion branches: tapped WMMA GEMMs with bn+gelu, summed
    launch_gemm(stream, mi, mb_w3, mb_s3, mb_b3, motA, M, 256, 768, 3,
                64, 64L * 768, 768, 1, M, 0, 256, 1, 1.0f, 1, 0);
    launch_gemm(stream, mi, mb_w5, mb_s5, mb_b5, motA, M, 256, 768, 5,
                64, 64L * 768, 768, 1, M, 0, 256, 1, 1.0f, 1, 1);
    launch_gemm(stream, mi, mb_w7, mb_s7, mb_b7, motA, M, 256, 768, 7,
                64, 64L * 768, 768, 1, M, 0, 256, 1, 1.0f, 1, 1);
    // 5) motion fuse (alpha = 1/3)
    launch_gemm(stream, motA, mf_w, mf_s, mf_b, motion, M, 256, 256, 1,
                M, 0, 256, 1, M, 0, 256, 1, 1.0f / 3.0f, 1, 0);
    // 6) DFT features
    zero_kernel<<<(1536 + 255) / 256, 256, 0, stream>>>(fe, 1536); // fe,sre,sim contiguous
    dft_kernel<<<NB * C_CH, 64, 0, stream>>>(dnT, fe, sre, sim);
    peak_kernel<<<NB, 64, 0, stream>>>(fe, sre, sim, pf, nst);
    // 7) phase gate MLP
    launch_gemm(stream, pf, pg_w1, pg_s1, pg_b1, pgh, M, 64, 3, 1,
                M, 0, 3, 1, M, 0, 64, 1, 1.0f, 1, 0);
    launch_gemm(stream, pgh, pg_w2, nullptr, pg_b2, pgate, M, 256, 64, 1,
                M, 0, 64, 1, M, 0, 256, 1, 1.0f, 2, 0);
    // 8) phase branch
    mul_kernel<<<M, 256, 0, stream>>>(dnT, pgate, dnp);
    launch_gemm(stream, dnp, pp_w, pp_s, pp_b, phase, M, 256, 256, 1,
                M, 0, 256, 1, M, 0, 256, 1, 1.0f, 1, 0);
    // 9) memory gate + scans
    launch_gemm(stream, gi, mg_w, nullptr, mg_b, gate, M, 256, 512, 1,
                M, 0, 512, 1, M, 0, 256, 1, 1.0f, 2, 0);
    scan_kernel<<<NB, 256, 0, stream>>>(dnT, gate, cat3);
    launch_gemm(stream, cat3, mo_w1, mo_s1, mo_b1, memh, M, 256, 768, 1,
                M, 0, 768, 1, M, 0, 256, 1, 1.0f, 1, 0);
    launch_gemm(stream, memh, mo_w2, nullptr, nullptr, memory, M, 256, 256, 1,
                M, 0, 256, 1, M, 0, 256, 1, 1.0f, 0, 0);
    // 10) router
    stats_router_kernel<<<NB, 256, 0, stream>>>(dnT, mi, motion, memory,
                                                rt_w1, rt_b1, rt_w2, rt_b2, nst);
    // 11) fuse + bottleneck
    cat4_kernel<<<M, 256, 0, stream>>>(dnT, motion, phase, memory, nst, cat4);
    launch_gemm(stream, cat4, bp_w1, bp_s1, bp_b1, bph, M, 256, 1024, 1,
                M, 0, 1024, 1, M, 0, 256, 1, 1.0f, 1, 0);
    launch_gemm(stream, bph, bp_w2, nullptr, nullptr, delta0, M, 256, 256, 1,
                M, 0, 256, 1, M, 0, 256, 1, 1.0f, 0, 0);
    // 12) feature gate, delta tokens
    launch_gemm(stream, dnT, fg_w, nullptr, fg_b, fgate, M, 256, 256, 1,
                M, 0, 256, 1, M, 0, 256, 1, 1.0f, 2, 0);
    dtok_kernel<<<M, 256, 0, stream>>>(delta0, fgate, res_s, dtok);
    // 13) spatial gate: strided-A GEMM over sp, output transposed to [n][c][hw]
    launch_gemm(stream, sp, sg_w, sg_s, sg_b, sgate, NB * HW, 256, 256, 1,
                HW, (long)C_CH * HW, 1, HW, HW, (long)C_CH * HW, 1, HW, 1.0f, 3, 0);
    // 14) broadcast add
    final_kernel<<<NB * C_CH * SEQ, 256, 0, stream>>>(x, dtok, sgate, out);
}